// SkipTranscoder_31293131718913
// MI455X (gfx1250) — compile-verified
//
#include <hip/hip_runtime.h>
#include <stdint.h>

#define BATCH 8192
#define DIN   1280
#define DOUT  1280
#define HDIM  16384
#define TOPK  32
#define NB_RED 1024

typedef float v2f __attribute__((ext_vector_type(2)));
typedef float v8f __attribute__((ext_vector_type(8)));

// ---- gfx1250 async global->LDS copy (ASYNCcnt-tracked), per-lane 16B ------
#if __has_builtin(__builtin_amdgcn_s_wait_asynccnt)
#define WAIT_ASYNCCNT(n) __builtin_amdgcn_s_wait_asynccnt(n)
#else
#define WAIT_ASYNCCNT(n) asm volatile("s_wait_asynccnt %0" ::"i"(n) : "memory")
#endif

__device__ __forceinline__ void async_copy_b128(const float* gsrc, void* ldst) {
  // Generic pointer to LDS: low 32 bits are the wave-relative LDS byte
  // address (flat LDS aperture, ISA 10.2).
  const unsigned loff = (unsigned)(size_t)ldst;
  asm volatile("global_load_async_to_lds_b128 %0, %1, off"
               :
               : "v"(loff), "v"(gsrc)
               : "memory");
}

// ---------------------------------------------------------------------------
// K1: C[M,N] = A[M,Kd] @ W[Kd,N] + bias[N]  via V_WMMA_F32_16X16X4_F32.
// 256 threads = 8 waves (2x4). Block tile 64(M) x 128(N), K-chunk 16.
// Each wave computes a 32x32 tile = 2x2 WMMA subtiles.
// Triple-buffered LDS, async global->LDS copies, 1 barrier/chunk.
// Fragment layouts per CDNA5 ISA 7.12.2:
//   A 16x4:  lane m=l&15, h=l>>4; vgpr0=K(2h), vgpr1=K(2h+1)
//   B 4x16:  lane n=l&15, h=l>>4; vgpr0=row K(2h) col n, vgpr1=row K(2h+1)
//   C 16x16: vgpr i holds row (h*8+i), col (l&15)
// Requires: M % 64 == 0, N % 128 == 0, Kd % 16 == 0.
// ---------------------------------------------------------------------------
__global__ void __launch_bounds__(256)
wmma_gemm_bias(const float* __restrict__ A, const float* __restrict__ W,
               const float* __restrict__ bias, float* __restrict__ C,
               int Kd, int N) {
  __shared__ float As[3][64][20];   // [buf][m][k] k-stride padded 16->20
  __shared__ float Ws[3][16][128];  // [buf][k][n]

  const int tid  = threadIdx.x;
  const int lane = tid & 31;
  const int wave = tid >> 5;
  const int wm   = wave >> 2;        // 0..1  (M direction)
  const int wn   = wave & 3;         // 0..3  (N direction)
  const int m    = lane & 15;
  const int h    = lane >> 4;

  const int m0 = blockIdx.y << 6;    // 64 rows per block
  const int n0 = blockIdx.x << 7;    // 128 cols per block

  // staging assignments: A tile 64x16 -> 1 x b128/thread; W tile 16x128 -> 2
  const int mp  = tid >> 2;          // 0..63
  const int kq  = (tid & 3) << 2;    // 0,4,8,12
  const int kr0 = tid >> 5;          // 0..7
  const int kr1 = kr0 + 8;           // 8..15
  const int c40 = (tid & 31) << 2;   // 0..124

  const float* gA  = A + (size_t)(m0 + mp) * Kd + kq;
  const float* gW0 = W + (size_t)kr0 * N + n0 + c40;
  const float* gW1 = W + (size_t)kr1 * N + n0 + c40;

  v8f acc[2][2] = {};
  const int NC = Kd >> 4;

  // prologue: issue chunks 0 and 1
  async_copy_b128(gA,  &As[0][mp][kq]);
  async_copy_b128(gW0, &Ws[0][kr0][c40]);
  async_copy_b128(gW1, &Ws[0][kr1][c40]);
  if (NC > 1) {
    async_copy_b128(gA + 16,             &As[1][mp][kq]);
    async_copy_b128(gW0 + (size_t)16 * N, &Ws[1][kr0][c40]);
    async_copy_b128(gW1 + (size_t)16 * N, &Ws[1][kr1][c40]);
    WAIT_ASYNCCNT(3);  // chunk 0 landed; chunk 1 may stay in flight
  } else {
    WAIT_ASYNCCNT(0);
  }
  __syncthreads();

  // fetch pointers for chunk c+2
  const float* gAf  = gA + 32;
  const float* gW0f = gW0 + (size_t)32 * N;
  const float* gW1f = gW1 + (size_t)32 * N;

  int buf  = 0;  // buffer holding chunk c
  int ibuf = 2;  // buffer receiving chunk c+2
  for (int c = 0; c < NC; ++c) {
    if (c + 2 < NC) {  // issue chunk c+2 (overlaps this chunk's compute)
      async_copy_b128(gAf,  &As[ibuf][mp][kq]);
      async_copy_b128(gW0f, &Ws[ibuf][kr0][c40]);
      async_copy_b128(gW1f, &Ws[ibuf][kr1][c40]);
      gAf  += 16;
      gW0f += (size_t)16 * N;
      gW1f += (size_t)16 * N;
      ibuf = (ibuf == 2) ? 0 : ibuf + 1;
    }
#pragma unroll
    for (int kk = 0; kk < 16; kk += 4) {
      const int ks = kk + (h << 1);
      v2f a[2], b[2];
#pragma unroll
      for (int mi = 0; mi < 2; ++mi) {
        const float2 av =
            *(const float2*)&As[buf][(wm << 5) + (mi << 4) + m][ks];
        a[mi].x = av.x;
        a[mi].y = av.y;
      }
#pragma unroll
      for (int ni = 0; ni < 2; ++ni) {
        const int nn = (wn << 5) + (ni << 4) + m;
        b[ni].x = Ws[buf][ks][nn];
        b[ni].y = Ws[buf][ks + 1][nn];
      }
#pragma unroll
      for (int mi = 0; mi < 2; ++mi)
#pragma unroll
        for (int ni = 0; ni < 2; ++ni)
          acc[mi][ni] = __builtin_amdgcn_wmma_f32_16x16x4_f32(
              /*neg_a=*/false, a[mi], /*neg_b=*/false, b[ni],
              /*c_mod=*/(short)0, acc[mi][ni],
              /*reuse_a=*/false, /*reuse_b=*/false);
    }
    if (c + 1 < NC) {
      // async loads complete in order: allowing 3 outstanding still
      // guarantees chunk c+1 (issued before c+2) has landed in LDS.
      if (c + 2 < NC) WAIT_ASYNCCNT(3);
      else            WAIT_ASYNCCNT(0);
      __syncthreads();
      buf = (buf == 2) ? 0 : buf + 1;
    }
  }

#pragma unroll
  for (int mi = 0; mi < 2; ++mi) {
#pragma unroll
    for (int ni = 0; ni < 2; ++ni) {
      const int col = n0 + (wn << 5) + (ni << 4) + m;
      const float bv = bias[col];
#pragma unroll
      for (int i = 0; i < 8; ++i) {
        const int row = m0 + (wm << 5) + (mi << 4) + (h << 3) + i;
        C[(size_t)row * N + col] = acc[mi][ni][i] + bv;
      }
    }
  }
}

// ---------------------------------------------------------------------------
// K2: exact per-row top-32 via 32-pass radix select; deterministic tie-break
// by lowest index (matches jax.lax.top_k). One 256-thread block per row; each
// thread owns 64 contiguous elements kept in registers (full unroll).
// ---------------------------------------------------------------------------
__device__ __forceinline__ int block_reduce_sum_all(int v, int* sh) {
#pragma unroll
  for (int o = 16; o > 0; o >>= 1) v += __shfl_down(v, o, 32);
  const int wid  = threadIdx.x >> 5;
  const int lane = threadIdx.x & 31;
  if (lane == 0) sh[wid] = v;
  __syncthreads();
  if (threadIdx.x == 0) {
    int s = 0;
#pragma unroll
    for (int i = 0; i < 8; ++i) s += sh[i];
    sh[8] = s;
  }
  __syncthreads();
  const int r = sh[8];
  __syncthreads();
  return r;
}

__global__ void __launch_bounds__(256)
topk_scatter(float* __restrict__ hidden, float* __restrict__ ws_vals,
             int* __restrict__ ws_idx) {
  __shared__ int s_red[9];
  __shared__ int s_cnt[512];   // [0..255]=cntG  [256..511]=cntE
  __shared__ int s_pre[514];   // [0..256]=prefG(+total) [257..513]=prefE

  const int row  = blockIdx.x;
  const int tid  = threadIdx.x;
  const int base = tid * 64;
  float* rowp = hidden + (size_t)row * HDIM;

  unsigned key[64];
#pragma unroll
  for (int i = 0; i < 64; i += 4) {
    const float4 f = *reinterpret_cast<const float4*>(rowp + base + i);
    const unsigned b0 = __float_as_uint(f.x), b1 = __float_as_uint(f.y),
                   b2 = __float_as_uint(f.z), b3 = __float_as_uint(f.w);
    key[i + 0] = b0 ^ ((b0 & 0x80000000u) ? 0xFFFFFFFFu : 0x80000000u);
    key[i + 1] = b1 ^ ((b1 & 0x80000000u) ? 0xFFFFFFFFu : 0x80000000u);
    key[i + 2] = b2 ^ ((b2 & 0x80000000u) ? 0xFFFFFFFFu : 0x80000000u);
    key[i + 3] = b3 ^ ((b3 & 0x80000000u) ? 0xFFFFFFFFu : 0x80000000u);
  }

  unsigned prefix = 0;
  int remaining = TOPK;
  for (int bit = 31; bit >= 0; --bit) {
    const unsigned bitm   = 1u << bit;
    const unsigned maskHi = (bit < 31) ? (0xFFFFFFFFu << (bit + 1)) : 0u;
    const unsigned selm   = maskHi | bitm;
    const unsigned want   = prefix | bitm;
    int c = 0;
#pragma unroll
    for (int i = 0; i < 64; ++i) c += ((key[i] & selm) == want) ? 1 : 0;
    c = block_reduce_sum_all(c, s_red);
    if (c >= remaining) prefix |= bitm;
    else remaining -= c;
  }
  const unsigned T = prefix;

  int cg = 0, ce = 0;
#pragma unroll
  for (int i = 0; i < 64; ++i) {
    cg += (key[i] > T) ? 1 : 0;
    ce += (key[i] == T) ? 1 : 0;
  }
  s_cnt[tid] = cg;
  s_cnt[256 + tid] = ce;
  __syncthreads();
  if (tid == 0) {
    int a = 0, b = 0;
    for (int t = 0; t < 256; ++t) {
      s_pre[t] = a;        a += s_cnt[t];
      s_pre[257 + t] = b;  b += s_cnt[256 + t];
    }
    s_pre[256] = a;  // totalG == TOPK - remaining
  }
  __syncthreads();
  const int prefG = s_pre[tid], prefE = s_pre[257 + tid], totalG = s_pre[256];

  int g = 0, e = 0;
#pragma unroll
  for (int i = 0; i < 64; ++i) {
    const unsigned k = key[i];
    float out = 0.0f;
    int sel = 0, rank = 0;
    if (k > T) {
      rank = prefG + g; ++g; sel = 1;
    } else if (k == T) {
      const int er = prefE + e; ++e;
      if (er < remaining) { rank = totalG + er; sel = 1; }
    }
    if (sel) {
      const unsigned fb = (k & 0x80000000u) ? (k ^ 0x80000000u) : ~k;
      const float val = __uint_as_float(fb);
      out = val > 0.0f ? val : 0.0f;
      ws_vals[row * TOPK + rank] = out;
      ws_idx[row * TOPK + rank]  = base + i;
    }
    rowp[base + i] = out;
  }
}

// ---------------------------------------------------------------------------
// K3: predicted[row,:] += sum_j val_j * W_dec[idx_j,:] + b_dec
// (skip output + b_skip already in predicted). W_dec rows hit in L2 (84 MB).
// ---------------------------------------------------------------------------
__global__ void __launch_bounds__(256)
decoder_skip_add(const float* __restrict__ ws_vals, const int* __restrict__ ws_idx,
                 const float* __restrict__ W_dec, const float* __restrict__ b_dec,
                 float* __restrict__ predicted) {
  __shared__ float sv[TOPK];
  __shared__ int   si[TOPK];
  const int row = blockIdx.x;
  const int tid = threadIdx.x;
  if (tid < TOPK) {
    sv[tid] = ws_vals[row * TOPK + tid];
    si[tid] = ws_idx[row * TOPK + tid];
  }
  __syncthreads();

  float acc[5] = {0.f, 0.f, 0.f, 0.f, 0.f};
  for (int j = 0; j < TOPK; ++j) {
    const float v = sv[j];           // uniform across block
    if (v > 0.0f) {                  // zero contributions are exact zeros
      const float* wr = W_dec + (size_t)si[j] * DOUT;
#pragma unroll
      for (int c = 0; c < 5; ++c) acc[c] += v * wr[tid + (c << 8)];
    }
  }
#pragma unroll
  for (int c = 0; c < 5; ++c) {
    const int col = tid + (c << 8);
    predicted[(size_t)row * DOUT + col] += acc[c] + b_dec[col];
  }
}

// ---------------------------------------------------------------------------
// K4/K5: deterministic two-stage reductions for MSE and l0.
// ---------------------------------------------------------------------------
__global__ void __launch_bounds__(256)
loss_partials(const float* __restrict__ predicted, const float* __restrict__ target,
              const float* __restrict__ ws_vals,
              float* __restrict__ partial_sq, float* __restrict__ partial_l0) {
  const size_t n  = (size_t)BATCH * DOUT;
  const size_t nk = (size_t)BATCH * TOPK;
  const size_t stride = (size_t)gridDim.x * blockDim.x;
  float s = 0.0f, c = 0.0f;
  for (size_t i = blockIdx.x * (size_t)blockDim.x + threadIdx.x; i < n; i += stride) {
    const float d = predicted[i] - target[i];
    s += d * d;
  }
  for (size_t i = blockIdx.x * (size_t)blockDim.x + threadIdx.x; i < nk; i += stride)
    c += (ws_vals[i] > 0.0f) ? 1.0f : 0.0f;

  __shared__ float sh[16];
#pragma unroll
  for (int o = 16; o > 0; o >>= 1) {
    s += __shfl_down(s, o, 32);
    c += __shfl_down(c, o, 32);
  }
  const int wid = threadIdx.x >> 5, lane = threadIdx.x & 31;
  if (lane == 0) { sh[wid] = s; sh[8 + wid] = c; }
  __syncthreads();
  if (threadIdx.x == 0) {
    float ts = 0.f, tc = 0.f;
#pragma unroll
    for (int i = 0; i < 8; ++i) { ts += sh[i]; tc += sh[8 + i]; }
    partial_sq[blockIdx.x] = ts;
    partial_l0[blockIdx.x] = tc;
  }
}

__global__ void finalize_loss(const float* __restrict__ partial_sq,
                              const float* __restrict__ partial_l0,
                              float* __restrict__ scalars) {
  if (threadIdx.x == 0 && blockIdx.x == 0) {
    double sq = 0.0, l0 = 0.0;
    for (int i = 0; i < NB_RED; ++i) { sq += partial_sq[i]; l0 += partial_l0[i]; }
    const float recon = (float)(sq / ((double)BATCH * (double)DOUT));
    const float l0m   = (float)(l0 / (double)BATCH);
    scalars[0] = recon;  // loss
    scalars[1] = recon;  // reconstruction_loss
    scalars[2] = 0.0f;   // sparsity_loss
    scalars[3] = l0m;    // l0
  }
}

// ---------------------------------------------------------------------------
extern "C" void kernel_launch(void* const* d_in, const int* in_sizes, int n_in,
                              void* d_out, int out_size, void* d_ws, size_t ws_size,
                              hipStream_t stream) {
  (void)in_sizes; (void)n_in; (void)out_size; (void)ws_size;
  const float* mlp_input  = (const float*)d_in[0];
  const float* mlp_output = (const float*)d_in[1];
  const float* W_enc  = (const float*)d_in[2];
  const float* b_enc  = (const float*)d_in[3];
  const float* W_dec  = (const float*)d_in[4];
  const float* b_dec  = (const float*)d_in[5];
  const float* W_skip = (const float*)d_in[6];
  const float* b_skip = (const float*)d_in[7];

  float* predicted = (float*)d_out;                       // [B, DOUT]
  float* hidden    = predicted + (size_t)BATCH * DOUT;    // [B, H]
  float* scalars   = hidden + (size_t)BATCH * HDIM;       // loss, recon, sparsity, l0

  float* ws_vals    = (float*)d_ws;                                   // [B, K]
  int*   ws_idx     = (int*)(ws_vals + (size_t)BATCH * TOPK);         // [B, K]
  float* partial_sq = (float*)(ws_idx + (size_t)BATCH * TOPK);        // [NB_RED]
  float* partial_l0 = partial_sq + NB_RED;                            // [NB_RED]

  // pre = X @ W_enc + b_enc  -> staged in hidden region of d_out
  wmma_gemm_bias<<<dim3(HDIM / 128, BATCH / 64), dim3(256), 0, stream>>>(
      mlp_input, W_enc, b_enc, hidden, DIN, HDIM);
  // skip = X @ W_skip + b_skip -> predicted
  wmma_gemm_bias<<<dim3(DOUT / 128, BATCH / 64), dim3(256), 0, stream>>>(
      mlp_input, W_skip, b_skip, predicted, DIN, DOUT);
  // exact top-32 + relu + scatter (overwrites pre with sparse hidden)
  topk_scatter<<<BATCH, 256, 0, stream>>>(hidden, ws_vals, ws_idx);
  // predicted += hidden @ W_dec + b_dec  (sparse gather form)
  decoder_skip_add<<<BATCH, 256, 0, stream>>>(ws_vals, ws_idx, W_dec, b_dec, predicted);
  // losses (deterministic two-stage reduction)
  loss_partials<<<NB_RED, 256, 0, stream>>>(predicted, mlp_output, ws_vals,
                                            partial_sq, partial_l0);
  finalize_loss<<<1, 32, 0, stream>>>(partial_sq, partial_l0, scalars);
}